// LossVariance_91036126806632
// MI455X (gfx1250) — compile-verified
//
#include <hip/hip_runtime.h>
#include <hip/hip_bf16.h>

// ---------------- problem constants (from reference) ----------------
#define BB   16
#define CC   19
#define HH   512
#define WW   512
#define NN   (HH * WW)       // 262144 pixels per batch
#define KK   64              // label bins
#define EPSF 1e-8f

// ---------------- tiling ----------------
#define PTILE   512          // pixels staged per chunk
#define PPAD    516          // LDS row stride (dwords): 16B-aligned rows, bank stride 4
#define CPB     8            // chunks per block -> 4096 px per block
#define NTHR    256          // 8 waves of 32
#define NCOL    40           // bin columns: 0-18 sum(c), 19 count, 20-38 sumsq(c), 39 pad
#define GROUPS  (PTILE / 4)  // 4-pixel groups per chunk

typedef __attribute__((ext_vector_type(2))) float v2f;
typedef __attribute__((ext_vector_type(8))) float v8f;

// Phase 1: segmented (count, sum, sumsq) via one-hot fp32 WMMA.
// bins layout: bins[b][k][NCOL]
__global__ __launch_bounds__(NTHR) void LossVar_phase1(
    const float* __restrict__ x,          // [B][C][H][W]
    const long long* __restrict__ tgt,    // [B][1][H][W] int64 labels 0..63
    float* __restrict__ bins)             // [B][K][NCOL], pre-zeroed
{
    __shared__ float vals[CC * PPAD];     // staged tile, channel-major rows
    __shared__ int   labs[PTILE];
    __shared__ float lbins[KK * NCOL];

    const int tid  = threadIdx.x;
    const int lane = tid & 31;
    const int wave = tid >> 5;            // 0..7
    const int b    = blockIdx.y;          // batch
    const int blk  = blockIdx.x;          // chunk-group within batch

    for (int i = tid; i < KK * NCOL; i += NTHR) lbins[i] = 0.0f;

    const v8f vzero = {0.f,0.f,0.f,0.f,0.f,0.f,0.f,0.f};
    v8f acc[4][3];
    #pragma unroll
    for (int lt = 0; lt < 4; ++lt)
        #pragma unroll
        for (int ct = 0; ct < 3; ++ct) acc[lt][ct] = vzero;

    // ---- per-lane invariants (hoisted channel selection: branchless hot loop) ----
    const int  m   = lane & 15;           // label-in-tile (A) / column (B,D)
    const int  hi  = lane >> 4;           // lane half: K rows 2*hi, 2*hi+1
    const int  base0 = m * PPAD;                                  // tile0: sum ch m
    const int  c1    = (m < 3) ? (16 + m) : ((m >= 4) ? (m - 4) : 0);
    const bool one1  = (m == 3);          // tile1 col 19 = count column (ones)
    const bool sq1   = (m >= 4);          // tile1 cols 20..31 = x^2 ch 0..11
    const int  base1 = c1 * PPAD;
    const bool ok2   = (m < 7);           // tile2 cols 32..38 = x^2 ch 12..18
    const int  base2 = (ok2 ? (12 + m) : 0) * PPAD;

    for (int chunk = 0; chunk < CPB; ++chunk) {
        const int n0 = (blk * CPB + chunk) * PTILE;
        __syncthreads();                  // previous tile fully consumed

        // stage labels (int64 -> int, values < 64)
        for (int p = tid; p < PTILE; p += NTHR)
            labs[p] = (int)tgt[(size_t)b * NN + n0 + p];

        // stage values: async global->LDS, 16B per lane-op (ASYNCcnt-tracked)
        {
            const float* xb = x + (size_t)b * CC * NN + n0;
            for (int i = tid; i < (CC * PTILE / 4); i += NTHR) {
                const int c  = i >> 7;            // float4-row index / 128
                const int p4 = (i & 127) << 2;    // float offset within row
                const float* gp = xb + (size_t)c * NN + p4;
                const unsigned lo = (unsigned)(uintptr_t)(&vals[c * PPAD + p4]);
                asm volatile("global_load_async_to_lds_b128 %0, %1, off"
                             :: "v"(lo), "v"((unsigned long long)(uintptr_t)gp)
                             : "memory");
            }
            asm volatile("s_wait_asynccnt 0" ::: "memory");
        }
        __syncthreads();

        // each wave owns groups g = wave, wave+8, ... (uniform per wave: EXEC all 1s at WMMA)
        #pragma unroll 2
        for (int g = wave; g < GROUPS; g += 8) {
            const int p0 = g * 4;
            const int4 lv = *(const int4*)&labs[p0];   // one ds_load_b128 broadcast
            const int la = hi ? lv.z : lv.x;  // label of pixel in K-row 2*hi   (A vgpr0)
            const int lb = hi ? lv.w : lv.y;  // label of pixel in K-row 2*hi+1 (A vgpr1)
            const int pr = p0 + hi * 2;       // pixel for B vgpr0 row; +1 for vgpr1 row

            const float t00 = vals[base0 + pr], t01 = vals[base0 + pr + 1];
            const float t10 = vals[base1 + pr], t11 = vals[base1 + pr + 1];
            const float t20 = vals[base2 + pr], t21 = vals[base2 + pr + 1];

            v2f Bt[3];
            Bt[0].x = t00;
            Bt[0].y = t01;
            const float u0 = sq1 ? t10 * t10 : t10;
            const float u1 = sq1 ? t11 * t11 : t11;
            Bt[1].x = one1 ? 1.0f : u0;
            Bt[1].y = one1 ? 1.0f : u1;
            Bt[2].x = ok2 ? t20 * t20 : 0.0f;
            Bt[2].y = ok2 ? t21 * t21 : 0.0f;

            #pragma unroll
            for (int lt = 0; lt < 4; ++lt) {
                const int kb = lt * 16 + m;
                v2f A;
                A.x = (la == kb) ? 1.0f : 0.0f;
                A.y = (lb == kb) ? 1.0f : 0.0f;
                #pragma unroll
                for (int ct = 0; ct < 3; ++ct) {
                    acc[lt][ct] = __builtin_amdgcn_wmma_f32_16x16x4_f32(
                        false, A, false, Bt[ct], (short)0, acc[lt][ct], false, false);
                }
            }
        }
    }

    __syncthreads();
    // D layout: vgpr v -> row M = v + 8*hi, col N = m
    #pragma unroll
    for (int lt = 0; lt < 4; ++lt) {
        #pragma unroll
        for (int ct = 0; ct < 3; ++ct) {
            const int col = ct * 16 + m;
            if (col < NCOL) {
                #pragma unroll
                for (int v = 0; v < 8; ++v) {
                    const int k = lt * 16 + v + hi * 8;
                    atomicAdd(&lbins[k * NCOL + col], acc[lt][ct][v]);
                }
            }
        }
    }
    __syncthreads();

    float* gb = bins + (size_t)b * KK * NCOL;
    for (int i = tid; i < KK * NCOL; i += NTHR)
        atomicAdd(&gb[i], lbins[i]);
}

// Phase 2: variance + loss reduction (tiny).
__global__ __launch_bounds__(NTHR) void LossVar_phase2(
    const float* __restrict__ bins, float* __restrict__ out)
{
    __shared__ float svar[BB];
    __shared__ float nun[BB];
    const int tid = threadIdx.x;
    if (tid < BB) { svar[tid] = 0.0f; nun[tid] = 0.0f; }
    __syncthreads();

    for (int pair = tid; pair < BB * KK; pair += NTHR) {
        const int b = pair >> 6, k = pair & 63;
        if (k == 0) continue;                      // label 0 = ignore
        const float* row = bins + ((size_t)b * KK + k) * NCOL;
        const float cnt = row[19];
        if (cnt > 0.0f) atomicAdd(&nun[b], 1.0f);
        if (cnt > 1.0f) {
            float a = 0.0f;
            #pragma unroll
            for (int c = 0; c < CC; ++c) {
                const float s = row[c], ss = row[20 + c];
                a += (ss - s * s / cnt) / (cnt - 1.0f);
            }
            atomicAdd(&svar[b], a);
        }
    }
    __syncthreads();
    if (tid == 0) {
        float t = 0.0f;
        for (int b = 0; b < BB; ++b) t += svar[b] / (nun[b] + EPSF);
        out[0] = t / (float)BB;
    }
}

extern "C" void kernel_launch(void* const* d_in, const int* in_sizes, int n_in,
                              void* d_out, int out_size, void* d_ws, size_t ws_size,
                              hipStream_t stream) {
    const float*     x   = (const float*)d_in[0];
    const long long* tgt = (const long long*)d_in[1];
    float* bins = (float*)d_ws;

    hipMemsetAsync(bins, 0, (size_t)BB * KK * NCOL * sizeof(float), stream);

    dim3 grid(NN / (PTILE * CPB), BB);   // 64 x 16 = 1024 blocks
    LossVar_phase1<<<grid, NTHR, 0, stream>>>(x, tgt, bins);
    LossVar_phase2<<<1, NTHR, 0, stream>>>(bins, (float*)d_out);
}